// KSoftDTW_32152125178367
// MI455X (gfx1250) — compile-verified
//
#include <hip/hip_runtime.h>

// Soft-DTW forward DP on CDNA5 (gfx1250, wave32):
// DPP16-pipelined strips + Tensor Data Mover (TDM) staging of the D stream.
//
//   R[i,j] = D[i,j] + softmin_1(R[i-1,j-1], R[i-1,j], R[i,j-1])
//   softmin(a,b,c) = m - log(exp(m-a)+exp(m-b)+exp(m-c)),  m = min(a,b,c)
//
// Latency-bound (64 MB traffic ~ 2.7us at 23.3 TB/s; the DP chain dominates).
// One workgroup (16 wave32s) per batch; 32 strips of 16 rows, strip k=tid>>4,
// lane=tid&15=row in strip. At strip-internal step s, lane i computes column
// s-i. Row-to-row propagation = one DPP16 row_shr:1 (VALU); its "invalid
// lanes keep old" semantics merges the LDS boundary value into head lanes
// (0,16) in the same instruction. Strips pipeline 2 phases apart, K=16
// steps/phase ((d-1)K=16 >= H-1=15), one barrier per phase (95 total).
// Boundary rows ride 64-entry LDS rings (producer lead < 33 cols).
//
// D feed: each strip's next 16x16 f32 window is DMA'd Global->LDS by
// TENSOR_LOAD_TO_LDS (one scalar-descriptor issue per strip per phase,
// TENSORcnt-tracked), double-buffered by phase parity (window parity == p
// parity for both half-strips). s_wait_tensorcnt gates residency,
// s_wait_dscnt gates buffer reuse; the DMA overlaps the barrier and the
// serial softmin of the current window. TDM ignores EXEC, so all issue
// logic runs on SGPR (readfirstlane-derived) conditions -> scalar branches.
// Falls back to direct global loads + global_prefetch_b8 if the TDM builtin
// is absent.

#define BIGV   100000000.0f
#define NN     512
#define KS     16                        // steps per phase
#define NSTRIP 32                        // 16-row strips
#define NPHASE (2 * (NSTRIP - 1) + 33)   // 62 + ceil(527/16) = 95
#define RING   64

#if __has_builtin(__builtin_amdgcn_tensor_load_to_lds)
#define TDM_STAGE 1
#else
#define TDM_STAGE 0
#endif

__device__ __forceinline__ void wait_tensorcnt0() {
#if __has_builtin(__builtin_amdgcn_s_wait_tensorcnt)
    __builtin_amdgcn_s_wait_tensorcnt(0);
#else
    asm volatile("s_wait_tensorcnt 0x0" ::: "memory");
#endif
}
__device__ __forceinline__ void wait_dscnt0() {
#if __has_builtin(__builtin_amdgcn_s_wait_dscnt)
    __builtin_amdgcn_s_wait_dscnt(0);
#else
    asm volatile("s_wait_dscnt 0x0" ::: "memory");
#endif
}

// u = (this lane is row 0 of its 16-lane strip) ? bv : r_m1 from lane-1.
__device__ __forceinline__ float shift_up16(float bv, float x) {
#if __has_builtin(__builtin_amdgcn_update_dpp)
    // DPP16 row_shr:1 — lanes 1..15 / 17..31 take lane-1's value; lanes 0,16
    // are out-of-row (invalid) and keep 'old' = bv (bound_ctrl=0).
    return __int_as_float(__builtin_amdgcn_update_dpp(
        __float_as_int(bv), __float_as_int(x),
        0x111 /*row_shr:1*/, 0xF, 0xF, false));
#else
    const float nb = __shfl_up(x, 1, 32);
    return ((threadIdx.x & 15) == 0) ? bv : nb;
#endif
}

#if TDM_STAGE
typedef unsigned int u32x4 __attribute__((ext_vector_type(4)));
typedef int          i32x4 __attribute__((ext_vector_type(4)));
typedef int          i32x8 __attribute__((ext_vector_type(8)));

__device__ __forceinline__ unsigned int lds_offset_of(const void* p) {
    return (unsigned int)(uintptr_t)(__attribute__((address_space(3))) const void*)p;
}

// One TDM descriptor: 16x16 f32 tile of a 512x512 f32 row-major tensor,
// packed contiguously (row-major) at LDS byte offset lds_off.
__device__ __forceinline__ void tdm_load_tile16x16(const float* gtile,
                                                   unsigned int lds_off) {
    const unsigned long long ga = (unsigned long long)(uintptr_t)gtile;
    u32x4 g0;
    g0.x = 1u;                                   // count=1 (valid), user mode
    g0.y = lds_off;                              // lds_addr (bytes)
    g0.z = (unsigned int)ga;                     // global_addr[31:0]
    g0.w = (unsigned int)((ga >> 32) & 0x01FFFFFFu) | 0x80000000u;  // [56:32] | type=2
    i32x8 g1;
    g1[0] = 0x00020000;   // workgroup_mask=0 (not in cluster), data_size=4B
    g1[1] = 0x02000000;   // atomic_barrier=0 | tensor_dim0=512 (low16)
    g1[2] = 0x02000000;   // tensor_dim0 hi=0 | tensor_dim1=512 (low16)
    g1[3] = 0x00100000;   // tensor_dim1 hi=0 | tile_dim0=16
    g1[4] = 0x00000010;   // tile_dim1=16, tile_dim2=0
    g1[5] = 0x00000200;   // tensor_dim0_stride=512 (low32)
    g1[6] = 0;            // stride0 hi=0 | tensor_dim1_stride low16 (=0)
    g1[7] = 4;            // tensor_dim1_stride = 4<<16 = 262144 = 512*512
    const i32x4 gz = {0, 0, 0, 0};
#if __clang_major__ >= 23
    const i32x8 gz8 = {0, 0, 0, 0, 0, 0, 0, 0};
    __builtin_amdgcn_tensor_load_to_lds(g0, g1, gz, gz, gz8, 0);
#else
    __builtin_amdgcn_tensor_load_to_lds(g0, g1, gz, gz, 0);
#endif
}
#endif

template <bool GUARD>
__device__ __forceinline__ void run_window(
    int sbase, int lane, int k, bool head, bool tail, bool phact,
    const float* __restrict__ dsrc,   // TDM: &dtile[k][0][lane][0]; else Dblk+row*NN
    float* __restrict__ Rblk, int rowoff,
    float (*ring)[RING], float& r_m1, float& up_prev) {
#pragma unroll
    for (int t = 0; t < KS; ++t) {
        const int col = sbase + t - lane;

        // Boundary value for head lanes (row 16k reads row 16k-1 via LDS ring).
        float bv = BIGV;
        if (head && k > 0) {
            if (!GUARD || (phact && col < NN))   // head lane: col = sbase+t >= 0 when phact
                bv = ring[k - 1][(sbase + t) & (RING - 1)];
        }

        // 'up' = R[row-1, col]; full-EXEC DPP (phase gating is wave-uniform).
        const float u = shift_up16(bv, r_m1);

        if (!GUARD || (phact && col >= 0 && col < NN)) {
            const float dg = up_prev;            // R[row-1, col-1]
            const float lf = r_m1;               // R[row,   col-1]
#if TDM_STAGE
            // Window wi lives in buffer wi&1 == (col>>4)&1; row stride 16 f32.
            const float d = dsrc[(((col >> 4) & 1) << 8) + (col & 15)];
#else
            const float d = dsrc[col];
#endif
            const float m = fminf(dg, fminf(u, lf));
            const float e = __expf(m - dg) + __expf(m - u) + __expf(m - lf);
            const float r = d + (m - __logf(e));

            Rblk[rowoff + col] = r;
            if (tail) ring[k][col & (RING - 1)] = r;
            r_m1 = r;
        }
        if (!GUARD || phact) up_prev = u;        // becomes diag next column
    }
}

__global__ __launch_bounds__(512, 1)
void softdtw_tdm_kernel(const float* __restrict__ D, float* __restrict__ R) {
    const int b    = blockIdx.x;
    const int tid  = threadIdx.x;
    const int k    = tid >> 4;          // strip id (0..31)
    const int lane = tid & 15;          // row within strip
    const int rowoff = tid * NN;        // row == tid
    // Scalar wave id: forces SGPR conditions (scalar branches) around TDM issue.
    const int swv  = __builtin_amdgcn_readfirstlane(tid) >> 5;

    const float* __restrict__ Dblk = D + (size_t)b * NN * NN;
    float*       __restrict__ Rblk = R + (size_t)b * NN * NN;

    __shared__ float ring[NSTRIP - 1][RING];     // 7.75 KB boundary rings
#if TDM_STAGE
    __shared__ float dtile[NSTRIP][2][16][16];   // 64 KB double-buffered D windows
    const float*       dsrc  = &dtile[k][0][lane][0];
    const unsigned int dbase = lds_offset_of(&dtile[0][0][0][0]);
#else
    const float* dsrc = Dblk + rowoff;
#endif

    float r_m1    = BIGV;                        // R[row, col-1]; BIG left border
    float up_prev = (tid == 0) ? 0.0f : BIGV;    // R[-1,-1] = 0, else BIG
    const bool head = (lane == 0);
    const bool tail = (lane == 15) && (k < NSTRIP - 1);

#if TDM_STAGE
    // Prime strip 0 / window 0 (its regular issue slot would be phase -1).
    if (swv == 0)
        tdm_load_tile16x16(Dblk, dbase);         // strip 0, window 0, buffer 0
#endif

    // Wave-uniform compute range: strips 2swv,2swv+1 compute in [4swv, 4swv+34].
    const int pmin = 4 * swv;
    const int pmax = 4 * swv + 34;

    for (int p = 0; p < NPHASE; ++p) {
        if (p >= pmin && p <= pmax) {
#if TDM_STAGE
            wait_tensorcnt0();                    // this phase's windows resident
#endif
            const int  sbase = (p - 2 * k) * KS;  // per half-strip
            const bool phact = (sbase >= 0) && (sbase <= NN + 14);
#if !TDM_STAGE
            {
                const int pcol = sbase + KS - lane;
                if (pcol >= 0 && pcol < NN)
                    __builtin_prefetch(dsrc + pcol, 0, 3);
            }
#endif
            // Fast path when BOTH half-strips of this wave are fully interior.
            const int s0 = (p - 4 * swv) * KS;
            if (s0 >= 48 && s0 <= 496) {
                run_window<false>(sbase, lane, k, head, tail, true,
                                  dsrc, Rblk, rowoff, ring, r_m1, up_prev);
            } else {
                run_window<true>(sbase, lane, k, head, tail, phact,
                                 dsrc, Rblk, rowoff, ring, r_m1, up_prev);
            }
#if TDM_STAGE
            // Stage next phase's windows into the buffer that held window wi-1
            // (dead after this phase). Retire this phase's LDS reads first; the
            // DMA then overlaps the barrier and the next window's compute.
            wait_dscnt0();
            {
                const int nb = (p + 1) & 1;
#pragma unroll
                for (int h = 0; h < 2; ++h) {
                    const int strip = 2 * swv + h;          // SGPR
                    const int wn    = p + 1 - 2 * strip;    // next window index
                    if (wn >= 0 && wn <= 31) {              // scalar branch
                        tdm_load_tile16x16(
                            Dblk + (strip * 16) * NN + wn * 16,
                            dbase + (unsigned int)(strip * 2 + nb) * 1024u);
                    }
                }
            }
#endif
        }
        __syncthreads();   // phase barrier: publishes boundary rings downstream
    }
}

extern "C" void kernel_launch(void* const* d_in, const int* in_sizes, int n_in,
                              void* d_out, int out_size, void* d_ws, size_t ws_size,
                              hipStream_t stream) {
    (void)in_sizes; (void)n_in; (void)d_ws; (void)ws_size; (void)out_size;
    const float* x   = (const float*)d_in[0];   // [32, 512, 512, 1] fp32
    float*       out = (float*)d_out;           // [32, 512, 512, 1] fp32

    dim3 grid(32);     // one workgroup per batch
    dim3 block(512);   // 16 wave32s = 32 half-strips of 16 rows
    softdtw_tdm_kernel<<<grid, block, 0, stream>>>(x, out);
}